// Attention_EVIT_21792664060029
// MI455X (gfx1250) — compile-verified
//
#include <hip/hip_runtime.h>
#include <math.h>

// ---- problem constants -----------------------------------------------------
#define B_    64
#define N_    197
#define C_    768
#define H_    12
#define D_    64
#define KEEP_ 137           // int(0.7 * 196)
#define NPAD  208           // 13 * 16 (padded key dimension)
#define SROW  210           // LDS row stride for scores -> conflict-free column reads
#define VROW  72            // LDS row stride for V tile (16B aligned, bank-friendly)

typedef float v2f __attribute__((ext_vector_type(2)));
typedef float v4f __attribute__((ext_vector_type(4)));
typedef float v8f __attribute__((ext_vector_type(8)));

__device__ __forceinline__ v8f zero8() {
  v8f z = {0.f, 0.f, 0.f, 0.f, 0.f, 0.f, 0.f, 0.f};
  return z;
}

// D = A(16x4,f32) * B(4x16,f32) + C(16x16,f32)
// A frag lane L -> row L%16, K = 2*(L/16)+j ; B frag lane L -> col L%16, K = 2*(L/16)+j
// C/D frag lane L, vgpr r -> row r + 8*(L/16), col L%16
__device__ __forceinline__ v8f wmma_f32(v2f a, v2f b, v8f c) {
  return __builtin_amdgcn_wmma_f32_16x16x4_f32(
      /*neg_a=*/false, a, /*neg_b=*/false, b,
      /*c_mod=*/(short)0, c, /*reuse_a=*/false, /*reuse_b=*/false);
}

// ---- shared 32x64-per-wave GEMM core: C = A[M,K] * W[Nout,K]^T --------------
// A row-major [M,K], W row-major [Nout,K]; K = 768. 8 WMMA per k-step vs 6 loads.
__device__ __forceinline__ void gemm_tile_768(const float* __restrict__ A,
                                              const float* __restrict__ W,
                                              int m_base, int n_base,
                                              int lane, v8f acc[2][4]) {
  const int hi = lane >> 4, lr = lane & 15;
  const float* ap0 = A + (size_t)(m_base + lr) * 768 + 2 * hi;
  const float* ap1 = ap0 + (size_t)16 * 768;
  const float* bp = W + (size_t)(n_base + lr) * 768 + 2 * hi;
#pragma unroll 2
  for (int k0 = 0; k0 < 768; k0 += 4) {
    v2f a0 = *(const v2f*)(ap0 + k0);
    v2f a1 = *(const v2f*)(ap1 + k0);
    v2f b0 = *(const v2f*)(bp + k0);
    v2f b1 = *(const v2f*)(bp + (size_t)16 * 768 + k0);
    v2f b2 = *(const v2f*)(bp + (size_t)32 * 768 + k0);
    v2f b3 = *(const v2f*)(bp + (size_t)48 * 768 + k0);
    acc[0][0] = wmma_f32(a0, b0, acc[0][0]);
    acc[0][1] = wmma_f32(a0, b1, acc[0][1]);
    acc[0][2] = wmma_f32(a0, b2, acc[0][2]);
    acc[0][3] = wmma_f32(a0, b3, acc[0][3]);
    acc[1][0] = wmma_f32(a1, b0, acc[1][0]);
    acc[1][1] = wmma_f32(a1, b1, acc[1][1]);
    acc[1][2] = wmma_f32(a1, b2, acc[1][2]);
    acc[1][3] = wmma_f32(a1, b3, acc[1][3]);
  }
}

// ---- kernel 1: QKV GEMM, scatter to [3][B][H][N][D] -------------------------
// grid (197, 9), block (32,8): block tile 64(M) x 256(N); M = B*N = 12608, Nout = 2304
__global__ __launch_bounds__(256) void qkv_gemm_kernel(
    const float* __restrict__ x, const float* __restrict__ qkv_w,
    const float* __restrict__ qkv_b, float* __restrict__ ws_qkv) {
  const int lane = threadIdx.x, w = threadIdx.y;
  const int wm = w >> 2, wn = w & 3;
  const int m_base = blockIdx.x * 64 + wm * 32;
  const int n_base = blockIdx.y * 256 + wn * 64;
  v8f acc[2][4] = {zero8(), zero8(), zero8(), zero8(),
                   zero8(), zero8(), zero8(), zero8()};
  gemm_tile_768(x, qkv_w, m_base, n_base, lane, acc);
  const int hi = lane >> 4, lr = lane & 15;
#pragma unroll
  for (int i = 0; i < 2; i++)
#pragma unroll
    for (int j = 0; j < 4; j++) {
      const int nn = n_base + j * 16 + lr;
      const float bias = qkv_b[nn];
      const int which = nn / C_;
      const int rem = nn - which * C_;
      const int h = rem >> 6, d = rem & 63;
#pragma unroll
      for (int r = 0; r < 8; r++) {
        const int mm = m_base + i * 16 + r + 8 * hi;
        const int bb = mm / N_;
        const int qq = mm - bb * N_;
        const size_t dst = ((((size_t)which * B_ + bb) * H_ + h) * N_ + qq) * D_ + d;
        ws_qkv[dst] = acc[i][j][r] + bias;
      }
    }
}

// ---- kernel 2: fused attention per (b, h, 16 query rows) --------------------
// grid (13, B*H), block (32,8)
__global__ __launch_bounds__(256) void attn_kernel(
    const float* __restrict__ ws_qkv, float* __restrict__ attn_out,
    float* __restrict__ cls_attn) {
  __shared__ float S[16 * SROW];
  __shared__ __align__(16) float Vs[NPAD * VROW];
  const int lane = threadIdx.x, w = threadIdx.y;
  const int hi = lane >> 4, lr = lane & 15;
  const int bh = blockIdx.y;
  const int b = bh / H_, h = bh % H_;
  const int m0 = blockIdx.x * 16;
  const size_t plane = (size_t)N_ * D_;
  const float* qp = ws_qkv + (((size_t)0 * B_ + b) * H_ + h) * plane;
  const float* kp = ws_qkv + (((size_t)1 * B_ + b) * H_ + h) * plane;
  const float* vp = ws_qkv + (((size_t)2 * B_ + b) * H_ + h) * plane;
  const float scale = 0.125f;  // D^-0.5 = 1/8

  // Phase 0: kick off async global->LDS staging of the (zero-padded) V tile.
  // 128b per transfer; overlaps with phases 1-2; consumed after asynccnt wait.
  {
    const int t = w * 32 + lane;
    for (int f = t; f < NPAD * 16; f += 256) {  // 16 float4 slots per V row
      const int r = f >> 4, c4 = (f & 15) * 4;
      if (r < N_) {
        const unsigned lds = (unsigned)(size_t)&Vs[r * VROW + c4];
        const unsigned long long ga =
            (unsigned long long)(size_t)(vp + (size_t)r * D_ + c4);
        asm volatile("global_load_async_to_lds_b128 %0, %1, off"
                     :: "v"(lds), "v"(ga) : "memory");
      } else {
        v4f z = {0.f, 0.f, 0.f, 0.f};
        *(v4f*)&Vs[r * VROW + c4] = z;  // zero-pad rows 197..207
      }
    }
  }

  // Phase 1: S = scale * Q Kt (keys tiled over waves), padded keys -> 0
  const int qrow = m0 + lr;
  const bool qvalid = qrow < N_;
  v2f aF[16];
#pragma unroll
  for (int kk = 0; kk < 16; kk++) {
    v2f a = {0.f, 0.f};
    if (qvalid) a = *(const v2f*)(qp + (size_t)qrow * D_ + kk * 4 + 2 * hi);
    aF[kk] = a;
  }
  for (int kt = w; kt < 13; kt += 8) {
    const int key = kt * 16 + lr;
    const bool kvalid = key < N_;
    v8f acc = zero8();
#pragma unroll
    for (int kk = 0; kk < 16; kk++) {
      v2f bF = {0.f, 0.f};
      if (kvalid) bF = *(const v2f*)(kp + (size_t)key * D_ + kk * 4 + 2 * hi);
      acc = wmma_f32(aF[kk], bF, acc);
    }
#pragma unroll
    for (int r = 0; r < 8; r++)
      S[(r + 8 * hi) * SROW + kt * 16 + lr] = acc[r] * scale;
  }
  __syncthreads();

  // Phase 2: row softmax (2 rows per wave), wave32 shuffle reductions
  for (int rr = 0; rr < 2; rr++) {
    const int row = w * 2 + rr;
    float* Srow = S + row * SROW;
    const int grow = m0 + row;
    if (grow >= N_) {
      for (int c = lane; c < NPAD; c += 32) Srow[c] = 0.f;
      continue;
    }
    float mx = -3.4e38f;
    for (int c = lane; c < N_; c += 32) mx = fmaxf(mx, Srow[c]);
    for (int off = 16; off > 0; off >>= 1) mx = fmaxf(mx, __shfl_xor(mx, off, 32));
    float sum = 0.f;
    for (int c = lane; c < NPAD; c += 32) {
      const float e = (c < N_) ? __expf(Srow[c] - mx) : 0.f;
      Srow[c] = e;
      sum += e;
    }
    for (int off = 16; off > 0; off >>= 1) sum += __shfl_xor(sum, off, 32);
    const float inv = 1.f / sum;
    for (int c = lane; c < NPAD; c += 32) Srow[c] *= inv;
    if (m0 == 0 && row == 0) {  // head-averaged cls attention over non-cls keys
      for (int c = lane; c < N_; c += 32)
        if (c >= 1)
          atomicAdd(&cls_attn[(size_t)b * (N_ - 1) + c - 1], Srow[c] * (1.f / (float)H_));
    }
  }
  // per-wave: all my async V transfers landed in LDS; barrier makes them
  // visible workgroup-wide.
  asm volatile("s_wait_asynccnt 0" ::: "memory");
  __syncthreads();

  // Phase 3: O = P V, both operands from LDS (waves 0..3 own 16-wide D slices)
  if (w < 4) {
    const int n0 = w * 16;
    v8f acc = zero8();
    for (int k0 = 0; k0 < NPAD; k0 += 4) {
      const int kk = k0 + 2 * hi;
      v2f aP;
      aP.x = S[lr * SROW + kk];
      aP.y = S[lr * SROW + kk + 1];
      v2f bP;
      bP.x = Vs[kk * VROW + n0 + lr];
      bP.y = Vs[(kk + 1) * VROW + n0 + lr];
      acc = wmma_f32(aP, bP, acc);
    }
#pragma unroll
    for (int r = 0; r < 8; r++) {
      const int mm = m0 + r + 8 * hi;
      if (mm < N_)
        attn_out[((size_t)b * N_ + mm) * C_ + h * D_ + n0 + lr] = acc[r];
    }
  }
}

// ---- kernel 3: output projection -------------------------------------------
// grid (197, 3), block (32,8)
__global__ __launch_bounds__(256) void proj_gemm_kernel(
    const float* __restrict__ a, const float* __restrict__ proj_w,
    const float* __restrict__ proj_b, float* __restrict__ out) {
  const int lane = threadIdx.x, w = threadIdx.y;
  const int wm = w >> 2, wn = w & 3;
  const int m_base = blockIdx.x * 64 + wm * 32;
  const int n_base = blockIdx.y * 256 + wn * 64;
  v8f acc[2][4] = {zero8(), zero8(), zero8(), zero8(),
                   zero8(), zero8(), zero8(), zero8()};
  gemm_tile_768(a, proj_w, m_base, n_base, lane, acc);
  const int hi = lane >> 4, lr = lane & 15;
#pragma unroll
  for (int i = 0; i < 2; i++)
#pragma unroll
    for (int j = 0; j < 4; j++) {
      const int nn = n_base + j * 16 + lr;
      const float bias = proj_b[nn];
#pragma unroll
      for (int r = 0; r < 8; r++) {
        const int mm = m_base + i * 16 + r + 8 * hi;
        out[(size_t)mm * C_ + nn] = acc[i][j][r] + bias;
      }
    }
}

// ---- kernel 0: zero the cls-attention accumulator ---------------------------
__global__ void zero_kernel(float* __restrict__ p, int n) {
  const int i = blockIdx.x * blockDim.x + threadIdx.x;
  if (i < n) p[i] = 0.f;
}

// ---- kernel 4: stable top-k by rank counting (matches jax.lax.top_k) --------
__global__ __launch_bounds__(256) void topk_kernel(
    const float* __restrict__ cls, float* __restrict__ idx_out,
    float* __restrict__ index_out) {
  __shared__ float vals[N_ - 1];
  const int b = blockIdx.x, t = threadIdx.x;
  if (t < N_ - 1) vals[t] = cls[(size_t)b * (N_ - 1) + t];
  __syncthreads();
  if (t < N_ - 1) {
    const float v = vals[t];
    int rank = 0;
    for (int j = 0; j < N_ - 1; j++) {
      const float fv = vals[j];
      rank += ((fv > v) || (fv == v && j < t)) ? 1 : 0;
    }
    if (rank < KEEP_) {
      const float tv = (float)t;
      idx_out[(size_t)b * KEEP_ + rank] = tv;
      float* rowp = index_out + ((size_t)b * KEEP_ + rank) * C_;
      for (int c = 0; c < C_; c++) rowp[c] = tv;
    }
  }
}

// ---- launcher ---------------------------------------------------------------
extern "C" void kernel_launch(void* const* d_in, const int* in_sizes, int n_in,
                              void* d_out, int out_size, void* d_ws, size_t ws_size,
                              hipStream_t stream) {
  const float* x      = (const float*)d_in[0];
  const float* qkv_w  = (const float*)d_in[1];
  const float* qkv_b  = (const float*)d_in[2];
  const float* proj_w = (const float*)d_in[3];
  const float* proj_b = (const float*)d_in[4];
  float* out = (float*)d_out;

  // output layout (flat f32, reference return order)
  const size_t OUT_OFF   = 0;                                    // [B,N,C]
  const size_t INDEX_OFF = (size_t)B_ * N_ * C_;                 // [B,KEEP,C]
  const size_t IDX_OFF   = INDEX_OFF + (size_t)B_ * KEEP_ * C_;  // [B,KEEP]
  const size_t CLS_OFF   = IDX_OFF + (size_t)B_ * KEEP_;         // [B,N-1]

  float* ws_qkv = (float*)d_ws;                         // [3][B][H][N][D]
  float* ws_ao  = ws_qkv + (size_t)3 * B_ * N_ * C_;    // [B,N,C] pre-proj
  float* cls    = out + CLS_OFF;

  hipLaunchKernelGGL(zero_kernel, dim3((B_ * (N_ - 1) + 255) / 256), dim3(256),
                     0, stream, cls, B_ * (N_ - 1));
  hipLaunchKernelGGL(qkv_gemm_kernel, dim3(197, 9), dim3(32, 8), 0, stream,
                     x, qkv_w, qkv_b, ws_qkv);
  hipLaunchKernelGGL(attn_kernel, dim3(13, B_ * H_), dim3(32, 8), 0, stream,
                     ws_qkv, ws_ao, cls);
  hipLaunchKernelGGL(proj_gemm_kernel, dim3(197, 3), dim3(32, 8), 0, stream,
                     ws_ao, proj_w, proj_b, out + OUT_OFF);
  hipLaunchKernelGGL(topk_kernel, dim3(B_), dim3(256), 0, stream,
                     cls, out + IDX_OFF, out + INDEX_OFF);
}